// GCN_13657996001664
// MI455X (gfx1250) — compile-verified
//
#include <hip/hip_runtime.h>
#include <hip/hip_bf16.h>

typedef float v2f __attribute__((ext_vector_type(2)));
typedef float v8f __attribute__((ext_vector_type(8)));
typedef int   v4i __attribute__((vector_size(16)));

#define HID 128
#define NUM_CODES 32
#define GROUPS 3
#define BN_EPS 1e-5f

#define AS1 __attribute__((address_space(1)))
#define AS3 __attribute__((address_space(3)))

#if __has_builtin(__builtin_amdgcn_global_load_async_to_lds_b128)
#define HAVE_ASYNC_LDS 1
#else
#define HAVE_ASYNC_LDS 0
#endif

__device__ __forceinline__ void wait_async_zero() {
#if __has_builtin(__builtin_amdgcn_s_wait_asynccnt)
    __builtin_amdgcn_s_wait_asynccnt(0);
#else
    asm volatile("s_wait_asynccnt 0" ::: "memory");
#endif
}

// ---------------------------------------------------------------------------
// Zero a float region
// ---------------------------------------------------------------------------
__global__ void zero_f32(float* __restrict__ p, long long n) {
    long long i = (long long)blockIdx.x * blockDim.x + threadIdx.x;
    if (i < n) p[i] = 0.0f;
}

// ---------------------------------------------------------------------------
// H[N,128] = X[N,128] @ W[128,128]  via V_WMMA_F32_16X16X4_F32
// Block: 256 threads = 8 waves. Block tile: 32 rows x 64 cols.
//   wave w: row sub-tile rt = w>>2 (0/1), col sub-tile ct = w&3
// grid = (ceil(n/32), 2)   (blockIdx.y selects which 64-col half of W)
// Tiles staged into LDS via GLOBAL_LOAD_ASYNC_TO_LDS_B128 (ASYNCcnt tracked).
// ---------------------------------------------------------------------------
#define WSLD 72    // padded LDS stride for W tile (128 x 64)
#define XSLD 132   // padded LDS stride for X tile (32 x 128)

__global__ __launch_bounds__(256) void gemm_xw_wmma(
        const float* __restrict__ X, const float* __restrict__ W,
        float* __restrict__ H, int n) {
    __shared__ float Ws[HID * WSLD];   // 36,864 B
    __shared__ float Xs[32 * XSLD];    // 16,896 B

    const int tid  = threadIdx.x;
    const int row0 = blockIdx.x * 32;
    const int ncol0 = blockIdx.y * 64;

#if HAVE_ASYNC_LDS
    // Direct memory -> LDS (no VGPR staging), 16B per lane per issue.
    for (int i = tid * 4; i < HID * 64; i += 256 * 4) {
        int r = i >> 6, c = i & 63;
        __builtin_amdgcn_global_load_async_to_lds_b128(
            (AS1 v4i*)&W[r * HID + ncol0 + c], (AS3 v4i*)&Ws[r * WSLD + c], 0, 0);
    }
    for (int i = tid * 4; i < 32 * HID; i += 256 * 4) {
        int r = i >> 7, c = i & 127;
        int gr = row0 + r;
        if (gr < n) {
            __builtin_amdgcn_global_load_async_to_lds_b128(
                (AS1 v4i*)&X[(long long)gr * HID + c], (AS3 v4i*)&Xs[r * XSLD + c], 0, 0);
        } else {
            float4 z = {0.f, 0.f, 0.f, 0.f};
            *(float4*)&Xs[r * XSLD + c] = z;
        }
    }
    wait_async_zero();
#else
    for (int i = tid * 4; i < HID * 64; i += 256 * 4) {
        int r = i >> 6, c = i & 63;
        *(float4*)&Ws[r * WSLD + c] = *(const float4*)&W[r * HID + ncol0 + c];
    }
    for (int i = tid * 4; i < 32 * HID; i += 256 * 4) {
        int r = i >> 7, c = i & 127;
        int gr = row0 + r;
        float4 v = {0.f, 0.f, 0.f, 0.f};
        if (gr < n) v = *(const float4*)&X[(long long)gr * HID + c];
        *(float4*)&Xs[r * XSLD + c] = v;
    }
#endif
    __syncthreads();

    const int wave = tid >> 5;
    const int lane = tid & 31;
    const int rt = wave >> 2;            // 0..1
    const int ct = wave & 3;             // 0..3
    const int mloc = rt * 16 + (lane & 15);  // A row within Xs
    const int kq   = (lane >> 4) * 2;        // 0 or 2 (K sub-offset)
    const int nn   = lane & 15;              // B/D column within tile
    const int nbase = ct * 16;               // column base within Ws

    v8f acc = {};
    for (int k = 0; k < HID; k += 4) {
        v2f a, b;
        a.x = Xs[mloc * XSLD + k + kq];
        a.y = Xs[mloc * XSLD + k + kq + 1];
        b.x = Ws[(k + kq) * WSLD + nbase + nn];
        b.y = Ws[(k + kq + 1) * WSLD + nbase + nn];
#if __has_builtin(__builtin_amdgcn_wmma_f32_16x16x4_f32)
        acc = __builtin_amdgcn_wmma_f32_16x16x4_f32(
                false, a, false, b, (short)0, acc, false, false);
#else
        for (int r = 0; r < 8; ++r) {
            int mrow = rt * 16 + (lane >> 4) * 8 + r;
            float s = acc[r];
            s = fmaf(Xs[mrow * XSLD + k + 0], Ws[(k + 0) * WSLD + nbase + nn], s);
            s = fmaf(Xs[mrow * XSLD + k + 1], Ws[(k + 1) * WSLD + nbase + nn], s);
            s = fmaf(Xs[mrow * XSLD + k + 2], Ws[(k + 2) * WSLD + nbase + nn], s);
            s = fmaf(Xs[mrow * XSLD + k + 3], Ws[(k + 3) * WSLD + nbase + nn], s);
            acc[r] = s;
        }
#endif
    }

    // D layout: VGPR r -> row (lane>>4)*8 + r, col = lane&15
    const int rbase = rt * 16 + (lane >> 4) * 8;
    const int gc = ncol0 + nbase + nn;
    for (int r = 0; r < 8; ++r) {
        int gr = row0 + rbase + r;
        if (gr < n) H[(long long)gr * HID + gc] = acc[r];
    }
}

// ---------------------------------------------------------------------------
// Degree: deg[row[e]] += 1  (self-loops folded in later as +1)
// ---------------------------------------------------------------------------
__global__ void deg_kernel(const int* __restrict__ row, float* __restrict__ deg,
                           long long E) {
    long long e = (long long)blockIdx.x * blockDim.x + threadIdx.x;
    if (e < E) atomicAdd(&deg[row[e]], 1.0f);
}

__global__ void dinv_kernel(float* __restrict__ deg, int n) {
    int i = blockIdx.x * blockDim.x + threadIdx.x;
    if (i < n) deg[i] = rsqrtf(deg[i] + 1.0f);   // in-place deg -> dinv
}

// ---------------------------------------------------------------------------
// Self-loop init: out[i,f] = H[i,f] * dinv[i]^2 + bias[f]
// ---------------------------------------------------------------------------
__global__ void agg_init(const float* __restrict__ H, const float* __restrict__ dinv,
                         const float* __restrict__ bias, float* __restrict__ out,
                         int n) {
    long long idx = (long long)blockIdx.x * blockDim.x + threadIdx.x;
    if (idx >= (long long)n * HID) return;
    int node = (int)(idx >> 7), f = (int)(idx & 127);
    float d = dinv[node];
    out[idx] = H[idx] * d * d + bias[f];
}

// ---------------------------------------------------------------------------
// Edge scatter: out[row] += H[col] * dinv[row]*dinv[col].  One wave per edge,
// lane handles 4 features. Prefetch the gather row of a lookahead edge into
// the WGP cache (global_prefetch_b8) to hide L2 latency on the random gather.
// ---------------------------------------------------------------------------
#define EDGE_LOOKAHEAD 2048
__global__ __launch_bounds__(256) void agg_edges(
        const float* __restrict__ H, const int* __restrict__ rowi,
        const int* __restrict__ coli, const float* __restrict__ dinv,
        float* __restrict__ out, long long E) {
    long long e = (long long)blockIdx.x * 8 + (threadIdx.x >> 5);
    if (e >= E) return;
    int lane = threadIdx.x & 31;

    long long ep = e + EDGE_LOOKAHEAD;
    if (ep < E) {
        int cp = coli[ep];
        __builtin_prefetch(&H[(long long)cp * HID + lane * 4], 0, 1);
    }

    int r = rowi[e], c = coli[e];
    float coef = dinv[r] * dinv[c];
    float4 hv = *(const float4*)&H[(long long)c * HID + lane * 4];
    float* o = &out[(long long)r * HID + lane * 4];
    atomicAdd(&o[0], hv.x * coef);
    atomicAdd(&o[1], hv.y * coef);
    atomicAdd(&o[2], hv.z * coef);
    atomicAdd(&o[3], hv.w * coef);
}

// ---------------------------------------------------------------------------
// BatchNorm statistics: per-feature sum and sum of squares (128 threads/block,
// thread = feature column, block strides 512 rows, one atomic per block).
// ---------------------------------------------------------------------------
#define BN_RPB 512
__global__ __launch_bounds__(128) void bn_stats(
        const float* __restrict__ x, float* __restrict__ sums,
        float* __restrict__ sumsq, int n) {
    int f = threadIdx.x;
    int r0 = blockIdx.x * BN_RPB;
    int r1 = min(r0 + BN_RPB, n);
    float s = 0.f, ss = 0.f;
    for (int r = r0; r < r1; ++r) {
        float v = x[(long long)r * HID + f];
        s += v; ss += v * v;
    }
    atomicAdd(&sums[f], s);
    atomicAdd(&sumsq[f], ss);
}

__global__ void bn_apply(float* __restrict__ x, const float* __restrict__ sums,
                         const float* __restrict__ sumsq,
                         const float* __restrict__ gamma,
                         const float* __restrict__ beta, int n) {
    long long idx = (long long)blockIdx.x * blockDim.x + threadIdx.x;
    if (idx >= (long long)n * HID) return;
    int f = (int)(idx & 127);
    float invn = 1.0f / (float)n;
    float mu = sums[f] * invn;
    float var = sumsq[f] * invn - mu * mu;
    float y = gamma[f] * (x[idx] - mu) * rsqrtf(var + BN_EPS) + beta[f];
    x[idx] = fmaxf(y, 0.0f);
}

// ---------------------------------------------------------------------------
// Residual VQ: one wave per node; lane j evaluates code j (32 codes = 32 lanes).
// Codebook transposed in LDS for conflict-free access. Forward-pass loss
// reduces to mean(final_residual^2).
// ---------------------------------------------------------------------------
__global__ __launch_bounds__(256) void rvq_kernel(
        const float* __restrict__ x, const float* __restrict__ cbs, // [3,32,128]
        float* __restrict__ ids_out,  // [N,6] as float; write cols colbase..+2
        int colbase, float* __restrict__ loss_acc, int n) {
    __shared__ float cbT[GROUPS * HID * NUM_CODES]; // transposed [g][k][code], 48KB
    __shared__ float cnorm[GROUPS * NUM_CODES];

    int tid = threadIdx.x;
    for (int i = tid; i < GROUPS * NUM_CODES * HID; i += 256) {
        int g = i / (NUM_CODES * HID);
        int rem = i - g * (NUM_CODES * HID);
        int code = rem >> 7, k = rem & 127;
        cbT[(g * HID + k) * NUM_CODES + code] = cbs[i];
    }
    __syncthreads();
    for (int i = tid; i < GROUPS * NUM_CODES; i += 256) {
        int g = i >> 5, code = i & 31;
        float s = 0.f;
        for (int k = 0; k < HID; ++k) {
            float v = cbT[(g * HID + k) * NUM_CODES + code];
            s = fmaf(v, v, s);
        }
        cnorm[i] = s;
    }
    __syncthreads();

    int lane = tid & 31;
    long long stride = (long long)gridDim.x * 8;
    for (long long node = (long long)blockIdx.x * 8 + (tid >> 5); node < n;
         node += stride) {
        const float* xr = &x[node * HID];
        float r0 = xr[lane], r1 = xr[lane + 32], r2 = xr[lane + 64], r3 = xr[lane + 96];
        for (int g = 0; g < GROUPS; ++g) {
            const float* cg = &cbT[g * HID * NUM_CODES];
            float acc = 0.f;
            #pragma unroll
            for (int q = 0; q < 4; ++q) {
                float rq = (q == 0) ? r0 : (q == 1) ? r1 : (q == 2) ? r2 : r3;
                for (int s = 0; s < 32; ++s) {
                    float rk = __shfl(rq, s, 32);          // broadcast resid[32q+s]
                    acc = fmaf(rk, cg[(q * 32 + s) * NUM_CODES + lane], acc);
                }
            }
            float d = cnorm[g * NUM_CODES + lane] - 2.0f * acc;  // ||r||^2 const
            // lexicographic butterfly argmin (matches jnp.argmin tie-break)
            int best = lane; float bd = d;
            #pragma unroll
            for (int off = 16; off; off >>= 1) {
                float od = __shfl_xor(bd, off, 32);
                int   ob = __shfl_xor(best, off, 32);
                if (od < bd || (od == bd && ob < best)) { bd = od; best = ob; }
            }
            r0 -= cg[(lane      ) * NUM_CODES + best];
            r1 -= cg[(lane +  32) * NUM_CODES + best];
            r2 -= cg[(lane +  64) * NUM_CODES + best];
            r3 -= cg[(lane +  96) * NUM_CODES + best];
            if (lane == 0) ids_out[node * 6 + colbase + g] = (float)best;
        }
        float sq = r0 * r0 + r1 * r1 + r2 * r2 + r3 * r3;
        #pragma unroll
        for (int off = 16; off; off >>= 1) sq += __shfl_xor(sq, off, 32);
        if (lane == 0) atomicAdd(loss_acc, sq);
    }
}

// ---------------------------------------------------------------------------
// Output projections: logits = x@lin_W + lin_b [N,40]; gnn = x@gnn_W + gnn_b [N,6]
// ---------------------------------------------------------------------------
__global__ void proj_kernel(const float* __restrict__ x,
                            const float* __restrict__ linW, const float* __restrict__ linb,
                            const float* __restrict__ gnnW, const float* __restrict__ gnnb,
                            float* __restrict__ out_logits, float* __restrict__ out_gnn,
                            int n) {
    long long t = (long long)blockIdx.x * blockDim.x + threadIdx.x;
    if (t >= (long long)n * 46) return;
    int c = (int)(t % 46);
    long long node = t / 46;
    const float* xr = &x[node * HID];
    if (c < 40) {
        float s = linb[c];
        for (int k = 0; k < HID; ++k) s = fmaf(xr[k], linW[k * 40 + c], s);
        out_logits[node * 40 + c] = s;
    } else {
        int cc = c - 40;
        float s = gnnb[cc];
        for (int k = 0; k < HID; ++k) s = fmaf(xr[k], gnnW[k * 6 + cc], s);
        out_gnn[node * 6 + cc] = s;
    }
}

__global__ void finalize_loss(const float* __restrict__ loss2,
                              float* __restrict__ out, float inv) {
    if (threadIdx.x == 0 && blockIdx.x == 0)
        out[0] = (loss2[0] + loss2[1]) * inv;
}

// ---------------------------------------------------------------------------
extern "C" void kernel_launch(void* const* d_in, const int* in_sizes, int n_in,
                              void* d_out, int out_size, void* d_ws, size_t ws_size,
                              hipStream_t stream) {
    const float* node_feat = (const float*)d_in[0];
    const int*   edge_idx  = (const int*)d_in[1];
    const float* W1        = (const float*)d_in[2];
    const float* b1        = (const float*)d_in[3];
    const float* W2        = (const float*)d_in[4];
    const float* b2        = (const float*)d_in[5];
    const float* bn_gamma  = (const float*)d_in[6];
    const float* bn_beta   = (const float*)d_in[7];
    const float* codebooks = (const float*)d_in[8];   // [2,3,32,128]
    const float* lin_W     = (const float*)d_in[9];
    const float* lin_b     = (const float*)d_in[10];
    const float* gnn_W     = (const float*)d_in[11];
    const float* gnn_b     = (const float*)d_in[12];

    const int N = in_sizes[0] / HID;
    const long long E = (long long)in_sizes[1] / 2;
    const int* row = edge_idx;
    const int* col = edge_idx + E;

    // workspace layout (floats)
    float* bufA   = (float*)d_ws;               // [N,128] gemm output
    float* bufB   = bufA + (long long)N * HID;  // [N,128] agg / bn'd features
    float* dinv   = bufB + (long long)N * HID;  // [N] deg -> dinv in place
    float* bnsum  = dinv + N;                   // [128]
    float* bnsq   = bnsum + HID;                // [128]
    float* lossac = bnsq + HID;                 // [2]
    const long long zcount = (long long)N + HID + HID + 2;

    // output layout (floats): [N*40 logits][1 loss][N*6 ids][N*6 gnn]
    float* out_logits = (float*)d_out;
    float* out_loss   = out_logits + (long long)N * 40;
    float* out_ids    = out_loss + 1;
    float* out_gnn    = out_ids + (long long)N * 6;

    const long long NH = (long long)N * HID;
    dim3 gemm_grid((N + 31) / 32, 2);

    // 0) zero accumulators (deg, bn sums, loss)
    zero_f32<<<(int)((zcount + 255) / 256), 256, 0, stream>>>(dinv, zcount);

    // ---- conv1 ----
    gemm_xw_wmma<<<gemm_grid, 256, 0, stream>>>(node_feat, W1, bufA, N);
    deg_kernel<<<(int)((E + 255) / 256), 256, 0, stream>>>(row, dinv, E);
    dinv_kernel<<<(N + 255) / 256, 256, 0, stream>>>(dinv, N);
    agg_init<<<(int)((NH + 255) / 256), 256, 0, stream>>>(bufA, dinv, b1, bufB, N);
    agg_edges<<<(int)((E + 7) / 8), 256, 0, stream>>>(bufA, row, col, dinv, bufB, E);

    // ---- batchnorm + relu ----
    bn_stats<<<(N + BN_RPB - 1) / BN_RPB, 128, 0, stream>>>(bufB, bnsum, bnsq, N);
    bn_apply<<<(int)((NH + 255) / 256), 256, 0, stream>>>(bufB, bnsum, bnsq,
                                                          bn_gamma, bn_beta, N);
    // ---- RVQ layer 0 ----
    rvq_kernel<<<(N + 7) / 8, 256, 0, stream>>>(bufB, codebooks, out_ids, 0,
                                                &lossac[0], N);
    // ---- conv2 ----
    gemm_xw_wmma<<<gemm_grid, 256, 0, stream>>>(bufB, W2, bufA, N);
    agg_init<<<(int)((NH + 255) / 256), 256, 0, stream>>>(bufA, dinv, b2, bufB, N);
    agg_edges<<<(int)((E + 7) / 8), 256, 0, stream>>>(bufA, row, col, dinv, bufB, E);

    // ---- RVQ layer 1 ----
    rvq_kernel<<<(N + 7) / 8, 256, 0, stream>>>(
        bufB, codebooks + GROUPS * NUM_CODES * HID, out_ids, 3, &lossac[1], N);

    // ---- projections + loss ----
    proj_kernel<<<(int)(((long long)N * 46 + 255) / 256), 256, 0, stream>>>(
        bufB, lin_W, lin_b, gnn_W, gnn_b, out_logits, out_gnn, N);
    finalize_loss<<<1, 32, 0, stream>>>(lossac, out_loss, 1.0f / ((float)N * HID));
}